// SrgcnSoftmaxHead_25744033972468
// MI455X (gfx1250) — compile-verified
//
#include <hip/hip_runtime.h>
#include <hip/hip_bf16.h>
#include <math.h>

// ---------------------------------------------------------------------------
// GAT-style forward (SrgcnSoftmaxHead):
//   h = x @ W                       -> WMMA f32 16x16x4 GEMM (LDS-staged A)
//   score_l = h.a_l ; score_r = h.a_r
//   e = leaky_relu(score_l[dst] + score_r[src])
//   val = segment_softmax(e, dst)
//   acc[dst] += val * h[src]        -> SPMM (L2-resident h gathers + atomics)
//   out = adaptive_relu(acc + bias, sigmoid((acc + bias)@fc + bf))
// ---------------------------------------------------------------------------

#define F_IN  256
#define F_OUT 128
#define ALPHA 0.2f

#define GEMM_ROWS  32          // rows of x per block (two 16-row M tiles)
#define LDS_STRIDE 260         // 256 + 4 pad: lane lr -> bank 4*lr (no 16-way conflict)

typedef __attribute__((ext_vector_type(2))) float v2f;
typedef __attribute__((ext_vector_type(8))) float v8f;

// ---------------- init: zero accumulators, -inf segmax ----------------------
__global__ __launch_bounds__(256) void init_kernel(float* __restrict__ accum,
                                                   size_t accum_n,
                                                   float* __restrict__ segmax,
                                                   float* __restrict__ segsum,
                                                   int n_nodes) {
    size_t i = (size_t)blockIdx.x * blockDim.x + threadIdx.x;
    if (i < accum_n) accum[i] = 0.0f;
    if (i < (size_t)n_nodes) {
        ((unsigned int*)segmax)[i] = 0xFF800000u;  // -inf
        segsum[i] = 0.0f;
    }
}

// ---------------- GEMM h = x @ W via V_WMMA_F32_16X16X4_F32 -----------------
// Block = 256 threads = 8 waves, covers 32 rows x 128 cols of h.
// 32x256 tile of x staged in LDS (coalesced float4 global loads, padded rows).
// Wave w owns cols [16w,16w+16) and computes TWO 16x16 C tiles (rows m0/m0+16)
// sharing each B fragment -> 2 WMMAs per B load pair, 128 WMMAs per wave.
__global__ __launch_bounds__(256) void gemm_h_kernel(const float* __restrict__ x,
                                                     const float* __restrict__ w,
                                                     float* __restrict__ h,
                                                     int n_nodes) {
    __shared__ float xs[GEMM_ROWS * LDS_STRIDE];

    const int tid  = threadIdx.x;
    const int wave = tid >> 5;              // 0..7 -> N tile
    const int lane = tid & 31;
    const int lr   = lane & 15;             // 0..15
    const int hi   = lane >> 4;             // 0 or 1 (lane half)
    const int m0   = blockIdx.x * GEMM_ROWS;
    const int n0   = wave * 16;
    if (m0 >= n_nodes) return;              // uniform per block

    // ---- stage x[m0 .. m0+32) rows into LDS: 8192 contiguous floats --------
    const float* xsrc = x + (size_t)m0 * F_IN;
    #pragma unroll
    for (int i = tid; i < GEMM_ROWS * F_IN / 4; i += 256) {
        float4 v = *(const float4*)(xsrc + 4 * i);
        int r = (4 * i) >> 8;               // / F_IN
        int c = (4 * i) & (F_IN - 1);
        *(float4*)(&xs[r * LDS_STRIDE + c]) = v;
    }
    __syncthreads();

    v8f c0 = {}, c1 = {};
    const float* a0p = &xs[lr * LDS_STRIDE];          // rows m0+lr
    const float* a1p = &xs[(16 + lr) * LDS_STRIDE];   // rows m0+16+lr

    #pragma unroll 4
    for (int k0 = 0; k0 < F_IN; k0 += 4) {
        // B 4x16: v0 = W[k0+2*hi][n], v1 = W[k0+2*hi+1][n]  (row-major W)
        v2f b;
        b.x = w[(size_t)(k0 + 2 * hi)     * F_OUT + n0 + lr];
        b.y = w[(size_t)(k0 + 2 * hi + 1) * F_OUT + n0 + lr];
        // A 16x4 fragments from LDS: lanes 0-15 K={0,1}, lanes 16-31 K={2,3}
        v2f a0 = *(const v2f*)(a0p + k0 + 2 * hi);
        v2f a1 = *(const v2f*)(a1p + k0 + 2 * hi);
        c0 = __builtin_amdgcn_wmma_f32_16x16x4_f32(false, a0, false, b,
                                                   (short)0, c0, false, false);
        c1 = __builtin_amdgcn_wmma_f32_16x16x4_f32(false, a1, false, b,
                                                   (short)0, c1, false, false);
    }

    // C/D layout: VGPR r -> M = r + 8*hi, N = lr
    #pragma unroll
    for (int r = 0; r < 8; ++r) {
        h[(size_t)(m0 + 8 * hi + r)      * F_OUT + n0 + lr] = c0[r];
        h[(size_t)(m0 + 16 + 8 * hi + r) * F_OUT + n0 + lr] = c1[r];
    }
}

// ---------------- per-node attention scores ---------------------------------
__global__ __launch_bounds__(256) void scores_kernel(const float* __restrict__ h,
                                                     const float* __restrict__ a_l,
                                                     const float* __restrict__ a_r,
                                                     float* __restrict__ sl,
                                                     float* __restrict__ sr,
                                                     int n_nodes) {
    int wid  = (int)(((size_t)blockIdx.x * blockDim.x + threadIdx.x) >> 5);
    int lane = threadIdx.x & 31;
    if (wid >= n_nodes) return;
    const float* hrow = h + (size_t)wid * F_OUT;
    float pl = 0.0f, pr = 0.0f;
    #pragma unroll
    for (int j = 0; j < 4; ++j) {
        float v = hrow[lane + 32 * j];
        pl += v * a_l[lane + 32 * j];
        pr += v * a_r[lane + 32 * j];
    }
    #pragma unroll
    for (int off = 16; off > 0; off >>= 1) {
        pl += __shfl_xor(pl, off, 32);
        pr += __shfl_xor(pr, off, 32);
    }
    if (lane == 0) { sl[wid] = pl; sr[wid] = pr; }
}

// ---------------- float atomic max via sign-split int encoding --------------
__device__ __forceinline__ void atomicMaxF32(float* addr, float v) {
    if (v >= 0.0f) atomicMax((int*)addr, __float_as_int(v));
    else           atomicMin((unsigned int*)addr, __float_as_uint(v));
}

// ---------------- edge pass 1: leaky-relu score + segment max ---------------
__global__ __launch_bounds__(256) void edge_score_kernel(const long long* __restrict__ row,
                                                         const long long* __restrict__ col,
                                                         const float* __restrict__ sl,
                                                         const float* __restrict__ sr,
                                                         float* __restrict__ ew,
                                                         float* __restrict__ segmax,
                                                         long long n_edges) {
    long long i = (long long)blockIdx.x * blockDim.x + threadIdx.x;
    if (i >= n_edges) return;
    int r = (int)row[i], c = (int)col[i];
    float e = sl[r] + sr[c];
    e = (e > 0.0f) ? e : ALPHA * e;
    ew[i] = e;
    atomicMaxF32(&segmax[r], e);
}

// ---------------- edge pass 2: exp(e - max) + segment sum -------------------
__global__ __launch_bounds__(256) void edge_exp_kernel(const long long* __restrict__ row,
                                                       float* __restrict__ ew,
                                                       const float* __restrict__ segmax,
                                                       float* __restrict__ segsum,
                                                       long long n_edges) {
    long long i = (long long)blockIdx.x * blockDim.x + threadIdx.x;
    if (i >= n_edges) return;
    int r = (int)row[i];
    float w = __expf(ew[i] - segmax[r]);
    ew[i] = w;
    atomicAdd(&segsum[r], w);
}

// ---------------- SPMM: acc[dst] += coeff * h[src] (one wave / edge) --------
__global__ __launch_bounds__(256) void spmm_kernel(const long long* __restrict__ row,
                                                   const long long* __restrict__ col,
                                                   const float* __restrict__ ew,
                                                   const float* __restrict__ segsum,
                                                   const float* __restrict__ h,
                                                   float* __restrict__ accum,
                                                   long long n_edges) {
    long long e = ((long long)blockIdx.x * blockDim.x + threadIdx.x) >> 5;
    int lane = threadIdx.x & 31;
    if (e >= n_edges) return;
    int r = (int)row[e], c = (int)col[e];
    float coeff = ew[e] / (segsum[r] + 1e-16f);
    const float4 hv = *(const float4*)(h + (size_t)c * F_OUT + 4 * lane);
    float* dst = accum + (size_t)r * F_OUT + 4 * lane;
    atomicAdd(dst + 0, coeff * hv.x);
    atomicAdd(dst + 1, coeff * hv.y);
    atomicAdd(dst + 2, coeff * hv.z);
    atomicAdd(dst + 3, coeff * hv.w);
}

// ---------------- finalize: bias, nan-scrub, gated relu ---------------------
__global__ __launch_bounds__(256) void finalize_kernel(const float* __restrict__ accum,
                                                       const float* __restrict__ bias,
                                                       const float* __restrict__ fc,
                                                       const float* __restrict__ bf,
                                                       float* __restrict__ out,
                                                       int n_nodes) {
    int wid  = (int)(((size_t)blockIdx.x * blockDim.x + threadIdx.x) >> 5);
    int lane = threadIdx.x & 31;
    if (wid >= n_nodes) return;

    float4 v  = *(const float4*)(accum + (size_t)wid * F_OUT + 4 * lane);
    // nan -> 0 before bias (matches reference order)
    if (v.x != v.x) v.x = 0.0f;
    if (v.y != v.y) v.y = 0.0f;
    if (v.z != v.z) v.z = 0.0f;
    if (v.w != v.w) v.w = 0.0f;
    const float4 b = *(const float4*)(bias + 4 * lane);
    v.x += b.x; v.y += b.y; v.z += b.z; v.w += b.w;

    const float4 f = *(const float4*)(fc + 4 * lane);
    float p = v.x * f.x + v.y * f.y + v.z * f.z + v.w * f.w;
    #pragma unroll
    for (int off = 16; off > 0; off >>= 1) p += __shfl_xor(p, off, 32);
    float gate = 1.0f / (1.0f + __expf(-(p + bf[0])));

    float4 o;
    o.x = (v.x < 0.0f ? 0.0f : v.x) + gate * (v.x > 0.0f ? 0.0f : v.x);
    o.y = (v.y < 0.0f ? 0.0f : v.y) + gate * (v.y > 0.0f ? 0.0f : v.y);
    o.z = (v.z < 0.0f ? 0.0f : v.z) + gate * (v.z > 0.0f ? 0.0f : v.z);
    o.w = (v.w < 0.0f ? 0.0f : v.w) + gate * (v.w > 0.0f ? 0.0f : v.w);
    *(float4*)(out + (size_t)wid * F_OUT + 4 * lane) = o;
}

// ---------------------------------------------------------------------------
extern "C" void kernel_launch(void* const* d_in, const int* in_sizes, int n_in,
                              void* d_out, int out_size, void* d_ws, size_t ws_size,
                              hipStream_t stream) {
    const float*     x      = (const float*)d_in[0];          // [N, 256]
    const long long* ei     = (const long long*)d_in[1];      // [2, E] int64
    /* edge_attr d_in[2] unused by reference */
    const float*     weight = (const float*)d_in[3];          // [256, 128]
    const float*     bias   = (const float*)d_in[4];          // [128]
    const float*     a_l    = (const float*)d_in[5];          // [128]
    const float*     a_r    = (const float*)d_in[6];          // [128]
    const float*     fc     = (const float*)d_in[7];          // [128,1]
    const float*     bf     = (const float*)d_in[8];          // [1]

    const int       N = in_sizes[0] / F_IN;
    const long long E = (long long)in_sizes[2];
    const long long* row = ei;          // dst
    const long long* col = ei + E;      // src

    // workspace layout (floats); total ~ 2*N*128 + 4*N + E  (~110 MB)
    float* ws      = (float*)d_ws;
    float* h       = ws;                               // N*128
    float* accum   = h + (size_t)N * F_OUT;            // N*128
    float* sl      = accum + (size_t)N * F_OUT;        // N
    float* sr      = sl + N;                           // N
    float* segmax  = sr + N;                           // N
    float* segsum  = segmax + N;                       // N
    float* ew      = segsum + N;                       // E

    const size_t accum_n = (size_t)N * F_OUT;

    // 1) init accumulators (deterministic each call)
    {
        int blocks = (int)((accum_n + 255) / 256);
        init_kernel<<<blocks, 256, 0, stream>>>(accum, accum_n, segmax, segsum, N);
    }
    // 2) h = x @ W  (WMMA f32, LDS-staged A, 2 C-tiles per wave)
    gemm_h_kernel<<<(N + GEMM_ROWS - 1) / GEMM_ROWS, 256, 0, stream>>>(x, weight, h, N);
    // 3) per-node scores
    scores_kernel<<<(N * 32 + 255) / 256, 256, 0, stream>>>(h, a_l, a_r, sl, sr, N);
    // 4) edge scores + segment max
    edge_score_kernel<<<(int)((E + 255) / 256), 256, 0, stream>>>(row, col, sl, sr, ew, segmax, E);
    // 5) exp + segment sum
    edge_exp_kernel<<<(int)((E + 255) / 256), 256, 0, stream>>>(row, ew, segmax, segsum, E);
    // 6) SPMM (one wave per edge)
    spmm_kernel<<<(int)((E * 32 + 255) / 256), 256, 0, stream>>>(row, col, ew, segsum, h, accum, E);
    // 7) finalize
    finalize_kernel<<<(N * 32 + 255) / 256, 256, 0, stream>>>(accum, bias, fc, bf, (float*)d_out, N);
}